// Net_56873956934085
// MI455X (gfx1250) — compile-verified
//
#include <hip/hip_runtime.h>
#include <hip/hip_bf16.h>
#include <math.h>

#define N_NODES 100000
#define N_EDGES 1600000
#define D_IN    128
#define D_OUT   64
#define AP      132            // padded LDS row stride (floats): banks 4*row mod 64 -> conflict-free

// Monotone float <-> uint mapping (radix-sort trick): unsigned max == float max.
#define FLIP_NEG_INF 0x007FFFFFu

__device__ __forceinline__ unsigned flipf(float f) {
    unsigned u = __float_as_uint(f);
    return (u & 0x80000000u) ? ~u : (u | 0x80000000u);
}
__device__ __forceinline__ float unflipf(unsigned u) {
    unsigned v = (u & 0x80000000u) ? (u & 0x7FFFFFFFu) : ~u;
    return __uint_as_float(v);
}

typedef __attribute__((ext_vector_type(2))) float v2f;
typedef __attribute__((ext_vector_type(8))) float v8f;
typedef __attribute__((ext_vector_type(4))) int   v4i;
typedef __attribute__((address_space(1))) v4i*    gv4i_p;   // global int4*
typedef __attribute__((address_space(3))) v4i*    lv4i_p;   // LDS int4*

// ---------------------------------------------------------------------------
// Kernel 1: init agg buffer to flipped(-inf)
// ---------------------------------------------------------------------------
__global__ void sage_init_agg(uint4* __restrict__ agg4) {
    int i = blockIdx.x * blockDim.x + threadIdx.x;   // exactly N*D_IN/4 threads
    agg4[i] = make_uint4(FLIP_NEG_INF, FLIP_NEG_INF, FLIP_NEG_INF, FLIP_NEG_INF);
}

// ---------------------------------------------------------------------------
// Kernel 2: edge scatter-max.  One wave32 per edge; lane l handles 4 floats.
// x row read is a coalesced 512B burst; x (51MB) is L2-resident on MI455X.
// global_atomic_max_u32 is order-independent -> bitwise deterministic.
// ---------------------------------------------------------------------------
__global__ void sage_scatter_max(const float* __restrict__ x,
                                 const long long* __restrict__ ei,
                                 unsigned* __restrict__ agg) {
    unsigned gtid = blockIdx.x * blockDim.x + threadIdx.x;
    unsigned e    = gtid >> 5;        // edge id (grid sized exactly to E*32)
    unsigned lane = gtid & 31u;       // 32 lanes x float4 = 128 features

    long long src = ei[e];
    long long dst = ei[(long long)N_EDGES + e];

    const float4* xr = (const float4*)(x + (size_t)src * D_IN);
    float4 v = xr[lane];

    unsigned* arow = agg + (size_t)dst * D_IN + lane * 4u;
    atomicMax(arow + 0, flipf(v.x));
    atomicMax(arow + 1, flipf(v.y));
    atomicMax(arow + 2, flipf(v.z));
    atomicMax(arow + 3, flipf(v.w));
}

// ---------------------------------------------------------------------------
// Kernel 3: fused  out = log_softmax(agg@Wl^T + b + x@Wr^T)
// 128 threads = 4 waves per 16-node tile. agg/x tiles staged into LDS once
// (CDNA5 async global->LDS copies, ASYNCcnt), agg un-flipped in place once,
// then each wave runs 64 chained V_WMMA_F32_16X16X4_F32 ops from ds_load_b64.
// ---------------------------------------------------------------------------
__global__ __launch_bounds__(128)
void sage_gemm_logsoftmax(const unsigned* __restrict__ agg,
                          const float* __restrict__ x,
                          const float* __restrict__ Wl,
                          const float* __restrict__ bl,
                          const float* __restrict__ Wr,
                          float* __restrict__ out) {
    __shared__ float ldsA[16 * AP];           // agg tile (flipped -> unflipped)
    __shared__ float ldsX[16 * AP];           // x tile
    __shared__ float tile[16][D_OUT + 4];
    __shared__ float rowmax[16];
    __shared__ float rowlse[16];

    const int tid  = threadIdx.x;
    const int lane = tid & 31;
    const int wave = tid >> 5;                // N-tile: classes [wave*16, +16)
    const int rc   = lane & 15;               // A row / B col within 16-tile
    const int hi   = lane >> 4;               // lane half select
    const int kb   = hi << 1;                 // K sub-offset: 0 or 2
    const int m0   = blockIdx.x * 16;         // node tile base (exact tiling)
    const int n0   = wave * 16;

    // ---- stage agg + x tiles (16 rows x 128 floats each) into LDS ----------
    const unsigned* gA = agg + (size_t)m0 * D_IN;
    const float*    gX = x   + (size_t)m0 * D_IN;

#if __has_builtin(__builtin_amdgcn_global_load_async_to_lds_b128)
    #pragma unroll
    for (int i = 0; i < 4; ++i) {             // 512 float4 chunks / 128 threads
        int idx  = tid + i * 128;             // 0..511
        int row  = idx >> 5;                  // 0..15
        int ch   = idx & 31;                  // 0..31 float4 chunks per row
        int goff = row * D_IN + ch * 4;
        int loff = row * AP + ch * 4;
        __builtin_amdgcn_global_load_async_to_lds_b128(
            (gv4i_p)(gA + goff), (lv4i_p)(ldsA + loff), 0, 0);
        __builtin_amdgcn_global_load_async_to_lds_b128(
            (gv4i_p)(gX + goff), (lv4i_p)(ldsX + loff), 0, 0);
    }
  #if __has_builtin(__builtin_amdgcn_s_wait_asynccnt)
    __builtin_amdgcn_s_wait_asynccnt(0);
  #else
    asm volatile("s_wait_asynccnt 0x0" ::: "memory");
  #endif
#else
    #pragma unroll
    for (int i = 0; i < 4; ++i) {
        int idx  = tid + i * 128;
        int row  = idx >> 5;
        int ch   = idx & 31;
        int goff = row * D_IN + ch * 4;
        int loff = row * AP + ch * 4;
        *(uint4*)(ldsA + loff)  = *(const uint4*)(gA + goff);
        *(float4*)(ldsX + loff) = *(const float4*)(gX + goff);
    }
#endif
    __syncthreads();

    // ---- un-flip + clamp(-inf -> 0) the agg tile in place, exactly once ----
    for (int i = tid; i < 16 * D_IN; i += 128) {
        int row = i >> 7, col = i & 127;
        unsigned u = __float_as_uint(ldsA[row * AP + col]);
        ldsA[row * AP + col] = (u == FLIP_NEG_INF) ? 0.0f : unflipf(u);
    }
    __syncthreads();

    // ---- WMMA: acc = aggTile @ Wl^T + xTile @ Wr^T  (K = 128 each) ---------
    const float* wlRow = Wl + (size_t)(n0 + rc) * D_IN;   // B[k][n] = Wl[n][k]
    const float* wrRow = Wr + (size_t)(n0 + rc) * D_IN;
    const float* la    = ldsA + rc * AP;
    const float* lx    = ldsX + rc * AP;

    v8f acc = {};
    #pragma unroll
    for (int k = 0; k < D_IN; k += 4) {
        v2f a = *(const v2f*)(la + k + kb);               // ds_load_b64
        v2f b; b.x = wlRow[k + kb]; b.y = wlRow[k + kb + 1];
        acc = __builtin_amdgcn_wmma_f32_16x16x4_f32(false, a, false, b,
                                                    (short)0, acc, false, false);
    }
    #pragma unroll
    for (int k = 0; k < D_IN; k += 4) {
        v2f a = *(const v2f*)(lx + k + kb);
        v2f b; b.x = wrRow[k + kb]; b.y = wrRow[k + kb + 1];
        acc = __builtin_amdgcn_wmma_f32_16x16x4_f32(false, a, false, b,
                                                    (short)0, acc, false, false);
    }

    // ---- + bias, stage 16x64 tile (C/D layout: M = r + 8*hi, N = lane&15) --
    const float bias = bl[n0 + rc];
    #pragma unroll
    for (int r = 0; r < 8; ++r) {
        int m = r + (hi << 3);
        tile[m][n0 + rc] = acc[r] + bias;
    }
    __syncthreads();

    // ---- row-wise log_softmax statistics (16 rows, one thread each) --------
    if (tid < 16) {
        const int m = tid;
        float mx = -INFINITY;
        #pragma unroll 8
        for (int j = 0; j < D_OUT; ++j) mx = fmaxf(mx, tile[m][j]);
        float s = 0.0f;
        #pragma unroll 8
        for (int j = 0; j < D_OUT; ++j) s += __expf(tile[m][j] - mx);
        rowmax[m] = mx;
        rowlse[m] = __logf(s);
    }
    __syncthreads();

    // ---- coalesced writeout: out[m][j] = v - max - log(sum(exp)) -----------
    for (int i = tid; i < 16 * D_OUT; i += 128) {
        int m = i >> 6;
        int j = i & 63;
        out[(size_t)(m0 + m) * D_OUT + j] = tile[m][j] - rowmax[m] - rowlse[m];
    }
}

// ---------------------------------------------------------------------------
extern "C" void kernel_launch(void* const* d_in, const int* in_sizes, int n_in,
                              void* d_out, int out_size, void* d_ws, size_t ws_size,
                              hipStream_t stream) {
    (void)in_sizes; (void)n_in; (void)out_size; (void)ws_size;

    const float*     x  = (const float*)d_in[0];
    const long long* ei = (const long long*)d_in[1];   // int64 edge_index [2,E]
    const float*     Wl = (const float*)d_in[2];
    const float*     bl = (const float*)d_in[3];
    const float*     Wr = (const float*)d_in[4];
    float*           out = (float*)d_out;

    unsigned* agg = (unsigned*)d_ws;                   // N_NODES * D_IN u32 (51.2MB)

    // 1) init agg = flipped(-inf)
    {
        const int total4 = N_NODES * D_IN / 4;         // 3,200,000 uint4
        sage_init_agg<<<total4 / 256, 256, 0, stream>>>((uint4*)agg);
    }
    // 2) scatter-max: one wave per edge
    {
        const long long threads = (long long)N_EDGES * 32;  // 51,200,000
        sage_scatter_max<<<(int)(threads / 256), 256, 0, stream>>>(x, ei, agg);
    }
    // 3) fused WMMA GEMM + bias + log_softmax  (100000 = 6250 * 16 exactly)
    {
        sage_gemm_logsoftmax<<<N_NODES / 16, 128, 0, stream>>>(agg, x, Wl, bl, Wr, out);
    }
}